// CaHTGP_17497696764445
// MI455X (gfx1250) — compile-verified
//
#include <hip/hip_runtime.h>
#include <math.h>

// ---------------------------------------------------------------------------
// Graph-attention GRU aggregation for MI455X (gfx1250, wave32).
// One wave = 16 sequences (WMMA M=16). All GEMMs via V_WMMA_F32_16X16X4_F32.
// ---------------------------------------------------------------------------

typedef __attribute__((ext_vector_type(2))) float v2f;
typedef __attribute__((ext_vector_type(8))) float v8f;

#define NWAVE 2      // waves per block
#define TSQ   16     // sequences per wave (WMMA M)
#define DIMH  64     // hidden dim D
#define TT    20     // neighbors / timesteps
#define LRA   0.2f   // LeakyReLU slope
#define NSEQ  (512 * 50)

static __device__ __forceinline__ v8f wmma4(v2f a, v2f b, v8f c) {
  // D = A(16x4 f32) * B(4x16 f32) + C(16x16 f32)
  return __builtin_amdgcn_wmma_f32_16x16x4_f32(
      /*neg_a=*/false, a, /*neg_b=*/false, b,
      /*c_mod=*/(short)0, c, /*reuse_a=*/false, /*reuse_b=*/false);
}

static __device__ __forceinline__ float sigmf(float x) {
  return 1.0f / (1.0f + __expf(-x));
}

__global__ __launch_bounds__(NWAVE * 32)
void CaHTGP_gruattn_kernel(const int* __restrict__ inputs,
                           const int* __restrict__ nbr_idx,
                           const int* __restrict__ nbr_msk,
                           const float* __restrict__ emb,
                           const float* __restrict__ W_ih,   // (192,128) row-major
                           const float* __restrict__ W_hh,   // (192,64)
                           const float* __restrict__ b_ih,   // (192)
                           const float* __restrict__ b_hh,   // (192)
                           const float* __restrict__ aw1,    // (64,128)
                           const float* __restrict__ ab1,    // (64)
                           const float* __restrict__ aw2,    // (64)
                           float* __restrict__ out)          // (25600,64)
{
  __shared__ float x_s[NWAVE][TSQ][128];      // [nb_t | central] per row
  __shared__ float h_s[NWAVE][2][TSQ][DIMH];  // GRU hidden, ping-pong
  __shared__ float sc_s[NWAVE][TSQ][TT];      // attention scores -> beta
  __shared__ int   nb_s[NWAVE][TSQ][TT];      // neighbor node ids
  __shared__ int   mk_s[NWAVE][TSQ][TT];      // neighbor masks
  __shared__ int   ms_s[NWAVE][TSQ];          // mask sums

  const int lane = threadIdx.x & 31;
  const int w    = threadIdx.x >> 5;
  const int seq0 = (blockIdx.x * NWAVE + w) * TSQ;

  // ---- stage indices, masks, zero score/h ---------------------------------
  for (int i = lane; i < TSQ * TT; i += 32) {
    const int r = i / TT, t = i % TT;
    const size_t g = (size_t)(seq0 + r) * TT + t;
    nb_s[w][r][t] = nbr_idx[g];
    mk_s[w][r][t] = nbr_msk[g];
    sc_s[w][r][t] = 0.0f;
  }
  for (int i = lane; i < TSQ * DIMH; i += 32)
    h_s[w][0][i >> 6][i & 63] = 0.0f;

  // ---- central embedding, L2-normalized, into x_s[.][.][64..127] ----------
  if (lane < TSQ) {
    const int idx = inputs[seq0 + lane];
    const float* e = emb + (size_t)idx * DIMH;
    float ss = 0.0f;
    for (int c = 0; c < DIMH; ++c) { const float v = e[c]; ss += v * v; }
    const float inv = 1.0f / fmaxf(sqrtf(ss), 1e-12f);
    for (int c = 0; c < DIMH; ++c) x_s[w][lane][DIMH + c] = e[c] * inv;
  }
  __syncthreads();

  const int nl = lane & 15;       // N column within tile / A row M
  const int kh = lane >> 4;       // K-half selector for A/B fragments
  const int m0 = kh * 8;          // C-layout M offset for this lane half

  int p = 0;                      // h ping-pong parity
  for (int t = 0; t < TT; ++t) {
    // -- stage nb_t rows (gather from embedding) into x_s[.][.][0..63] ------
    for (int i = lane; i < TSQ * 16; i += 32) {
      const int r = i >> 4, q = i & 15;
      const float4 v = *(const float4*)(emb + (size_t)nb_s[w][r][t] * DIMH + q * 4);
      *(float4*)&x_s[w][r][q * 4] = v;
      if (t + 1 < TT)
        __builtin_prefetch(emb + (size_t)nb_s[w][r][t + 1] * DIMH + q * 4, 0, 1);
    }
    __syncthreads();

    // -- GRU step: 4 column chunks of the 3 gates ---------------------------
    for (int j = 0; j < 4; ++j) {
      v8f ar = {}, az = {}, anx = {}, anh = {};

      // gates_x += x_t @ W_ih^T   (K = 128, A from LDS x_s)
      for (int k = 0; k < 128; k += 4) {
        const int kk = k + kh * 2;
        v2f a; a.x = x_s[w][nl][kk]; a.y = x_s[w][nl][kk + 1];
        const v2f br = *(const v2f*)(W_ih + (size_t)((j     ) * 16 + nl) * 128 + kk);
        const v2f bz = *(const v2f*)(W_ih + (size_t)((j +  4) * 16 + nl) * 128 + kk);
        const v2f bn = *(const v2f*)(W_ih + (size_t)((j +  8) * 16 + nl) * 128 + kk);
        ar  = wmma4(a, br, ar);
        az  = wmma4(a, bz, az);
        anx = wmma4(a, bn, anx);
      }
      // gates_h += h @ W_hh^T     (K = 64, A from LDS h_s[p])
      for (int k = 0; k < DIMH; k += 4) {
        const int kk = k + kh * 2;
        v2f a; a.x = h_s[w][p][nl][kk]; a.y = h_s[w][p][nl][kk + 1];
        const v2f br = *(const v2f*)(W_hh + (size_t)((j     ) * 16 + nl) * DIMH + kk);
        const v2f bz = *(const v2f*)(W_hh + (size_t)((j +  4) * 16 + nl) * DIMH + kk);
        const v2f bn = *(const v2f*)(W_hh + (size_t)((j +  8) * 16 + nl) * DIMH + kk);
        ar  = wmma4(a, br, ar);
        az  = wmma4(a, bz, az);
        anh = wmma4(a, bn, anh);
      }

      const int nc   = j * 16 + nl;
      const float br2  = b_ih[nc]        + b_hh[nc];
      const float bz2  = b_ih[64 + nc]   + b_hh[64 + nc];
      const float bnx  = b_ih[128 + nc];
      const float bnh  = b_hh[128 + nc];
#pragma unroll
      for (int v = 0; v < 8; ++v) {
        const int m = v + m0;
        const float r = sigmf(ar[v] + br2);
        const float z = sigmf(az[v] + bz2);
        const float n = tanhf(anx[v] + bnx + r * (anh[v] + bnh));
        const float hold = h_s[w][p][m][nc];
        h_s[w][1 - p][m][nc] = (1.0f - z) * n + z * hold;
      }
    }
    __syncthreads();
    p ^= 1;

    // -- attention: hid = leaky(a_in @ aw1^T + ab1); score += hid . aw2 -----
    for (int j = 0; j < 4; ++j) {
      v8f acc = {};
      for (int k = 0; k < 128; k += 4) {          // k<64 -> h_new, else central
        const int kk = k + kh * 2;
        v2f a;
        if (k < 64) { a.x = h_s[w][p][nl][kk]; a.y = h_s[w][p][nl][kk + 1]; }
        else        { a.x = x_s[w][nl][kk];    a.y = x_s[w][nl][kk + 1];    }
        const v2f b = *(const v2f*)(aw1 + (size_t)(j * 16 + nl) * 128 + kk);
        acc = wmma4(a, b, acc);
      }
      const float bb = ab1[j * 16 + nl];
      const float w2 = aw2[j * 16 + nl];
#pragma unroll
      for (int v = 0; v < 8; ++v) {
        float hv = acc[v] + bb;
        hv = (hv > 0.0f) ? hv : LRA * hv;
        atomicAdd(&sc_s[w][v + m0][t], hv * w2);
      }
    }
    __syncthreads();
  }

  // ---- masked softmax over neighbors (one lane per row) -------------------
  if (lane < TSQ) {
    int ms = 0;
    float mx = -1e30f;
    float s[TT];
    for (int t = 0; t < TT; ++t) {
      ms += mk_s[w][lane][t];
      s[t] = mk_s[w][lane][t] ? sc_s[w][lane][t] : -1e9f;
      mx = fmaxf(mx, s[t]);
    }
    float sum = 0.0f;
    for (int t = 0; t < TT; ++t) { s[t] = __expf(s[t] - mx); sum += s[t]; }
    const float inv = 1.0f / sum;
    for (int t = 0; t < TT; ++t) sc_s[w][lane][t] = s[t] * inv;
    ms_s[w][lane] = ms;
  }
  __syncthreads();

  // ---- z = sum_t beta_t * nb_t ; fallback to central if no neighbors ------
  {
    const int m  = lane >> 1;
    const int c0 = (lane & 1) * 32;
    float4 acc[8];
#pragma unroll
    for (int q = 0; q < 8; ++q) acc[q] = make_float4(0.f, 0.f, 0.f, 0.f);

    for (int t = 0; t < TT; ++t) {
      const float b = sc_s[w][m][t];
      const float* e = emb + (size_t)nb_s[w][m][t] * DIMH + c0;
#pragma unroll
      for (int q = 0; q < 8; ++q) {
        const float4 v = *(const float4*)(e + q * 4);
        acc[q].x += b * v.x; acc[q].y += b * v.y;
        acc[q].z += b * v.z; acc[q].w += b * v.w;
      }
    }
    const bool has = ms_s[w][m] > 0;
    float* op = out + (size_t)(seq0 + m) * DIMH + c0;
#pragma unroll
    for (int q = 0; q < 8; ++q) {
      float4 r = acc[q];
      if (!has) {
        r.x = x_s[w][m][DIMH + c0 + q * 4 + 0];
        r.y = x_s[w][m][DIMH + c0 + q * 4 + 1];
        r.z = x_s[w][m][DIMH + c0 + q * 4 + 2];
        r.w = x_s[w][m][DIMH + c0 + q * 4 + 3];
      }
      *(float4*)(op + q * 4) = r;
    }
  }
}

extern "C" void kernel_launch(void* const* d_in, const int* in_sizes, int n_in,
                              void* d_out, int out_size, void* d_ws, size_t ws_size,
                              hipStream_t stream) {
  const int*   inputs = (const int*)d_in[0];
  const int*   nseq   = (const int*)d_in[1];
  const int*   nmask  = (const int*)d_in[2];
  const float* emb    = (const float*)d_in[3];
  const float* W_ih   = (const float*)d_in[4];
  const float* W_hh   = (const float*)d_in[5];
  const float* b_ih   = (const float*)d_in[6];
  const float* b_hh   = (const float*)d_in[7];
  const float* aw1    = (const float*)d_in[8];
  const float* ab1    = (const float*)d_in[9];
  const float* aw2    = (const float*)d_in[10];
  float* outp = (float*)d_out;

  const int n_tiles = NSEQ / TSQ;          // 1600 wave tiles
  dim3 grid(n_tiles / NWAVE);              // 800 blocks
  dim3 block(NWAVE * 32);                  // 64 threads = 2 waves
  hipLaunchKernelGGL(CaHTGP_gruattn_kernel, grid, block, 0, stream,
                     inputs, nseq, nmask, emb, W_ih, W_hh, b_ih, b_hh,
                     aw1, ab1, aw2, outp);
}